// H100SmartEmbedding_63324997812722
// MI455X (gfx1250) — compile-verified
//
#include <hip/hip_runtime.h>
#include <hip/hip_bf16.h>
#include <stdint.h>

// [num_features, 768] f32 output. 384 MB of pure streaming writes; tables are
// 31 KB total. Store-bandwidth-bound: ~17 us floor at 23.3 TB/s.
//
// CDNA5 paths used:
//   - global_load_async_to_lds_b128 + s_wait_asynccnt (ASYNCcnt) to stage the
//     six embedding tables into LDS once per block.
//   - non-temporal global_store_b128 for the 384 MB output stream (output is
//     2x the 192 MB L2; NT avoids evicting everything else).

typedef float v4f __attribute__((ext_vector_type(4)));

#define ROW_F4   192u            // 768 floats per row = 192 float4
#define LUT_F4   1984u           // (1+1+4+8+16+32) rows * 32 float4 = 31744 B

// LDS layout (float4 units): price[0,32) size[32,64) exch[64,192)
// pair[192,448) level[448,960) time[960,1984)

__device__ __forceinline__ void async_copy_b128(unsigned lds_byte_off, uint64_t gaddr) {
  // GV mode: vdst = LDS byte offset (per-lane), vaddr = 64-bit global address.
  asm volatile("global_load_async_to_lds_b128 %0, %1, off"
               :: "v"(lds_byte_off), "v"(gaddr) : "memory");
}

__device__ __forceinline__ void wait_async0() {
#if defined(__has_builtin)
#if __has_builtin(__builtin_amdgcn_s_wait_asynccnt)
  __builtin_amdgcn_s_wait_asynccnt(0);
  return;
#endif
#endif
  asm volatile("s_wait_asynccnt 0" ::: "memory");
}

template <unsigned MOD, unsigned BASE>
__device__ __forceinline__ void stream_rows(const v4f* __restrict__ lut,
                                            v4f* __restrict__ out4,
                                            unsigned tid, unsigned row0,
                                            unsigned rstride, unsigned nrows) {
  const unsigned w = tid & 31u;
#pragma unroll 2
  for (unsigned row = row0; row < nrows; row += rstride) {
    // Compile-time modulus -> mul-hi sequence, no runtime division.
    const unsigned r = (MOD <= 1u) ? 0u : (row % MOD);
    v4f val = lut[BASE + r * 32u + w];              // conflict-free ds_load_b128
    __builtin_nontemporal_store(val, out4 + (size_t)row * ROW_F4 + tid);
  }
}

__global__ __launch_bounds__(192) void smart_embedding_kernel(
    const float* __restrict__ price_w, const float* __restrict__ size_w,
    const float* __restrict__ exchange_w, const float* __restrict__ pair_w,
    const float* __restrict__ level_w, const float* __restrict__ time_w,
    v4f* __restrict__ out4, unsigned nrows) {
  __shared__ v4f lut[LUT_F4];

  const unsigned tid = threadIdx.x;  // 0..191

  // ---- Stage all tables into LDS with async DMA (gfx1250 path) ----
  for (unsigned j = tid; j < LUT_F4; j += 192u) {
    const float* src;
    unsigned rel;  // float4 offset within the source array
    if (j < 32u)       { src = price_w;    rel = j; }
    else if (j < 64u)  { src = size_w;     rel = j - 32u; }
    else if (j < 192u) { src = exchange_w; rel = j - 64u; }
    else if (j < 448u) { src = pair_w;     rel = j - 192u; }
    else if (j < 960u) { src = level_w;    rel = j - 448u; }
    else               { src = time_w;     rel = j - 960u; }
    const unsigned lds_off = (unsigned)(uintptr_t)(&lut[j]);     // low 32 bits = LDS byte addr
    const uint64_t ga = (uint64_t)(uintptr_t)(src + 4u * rel);   // rel float4 -> floats
    async_copy_b128(lds_off, ga);
  }
  wait_async0();
  __syncthreads();

  // ---- Stream output: wave k (tid/32) owns one table type -> uniform branch ----
  const unsigned type = tid >> 5;
  const unsigned row0 = blockIdx.x;
  const unsigned rstride = gridDim.x;
  switch (type) {
    case 0: stream_rows<1u,   0u>(lut, out4, tid, row0, rstride, nrows); break;  // price (const)
    case 1: stream_rows<1u,  32u>(lut, out4, tid, row0, rstride, nrows); break;  // size  (const)
    case 2: stream_rows<3u,  64u>(lut, out4, tid, row0, rstride, nrows); break;  // exchange, i%3
    case 3: stream_rows<7u, 192u>(lut, out4, tid, row0, rstride, nrows); break;  // pair, i%7
    case 4: stream_rows<15u,448u>(lut, out4, tid, row0, rstride, nrows); break;  // level, i%15
    default:stream_rows<31u,960u>(lut, out4, tid, row0, rstride, nrows); break;  // time, i%31
  }
}

extern "C" void kernel_launch(void* const* d_in, const int* in_sizes, int n_in,
                              void* d_out, int out_size, void* d_ws, size_t ws_size,
                              hipStream_t stream) {
  (void)in_sizes; (void)n_in; (void)d_ws; (void)ws_size;
  const float* price_w    = (const float*)d_in[0];
  const float* size_w     = (const float*)d_in[1];
  const float* exchange_w = (const float*)d_in[2];
  const float* pair_w     = (const float*)d_in[3];
  const float* level_w    = (const float*)d_in[4];
  const float* time_w     = (const float*)d_in[5];
  // num_features (d_in[6]) lives in device memory; recover it host-side from
  // out_size so grid sizing is deterministic and graph-capture safe.
  const unsigned nrows = (unsigned)(out_size / 768);
  if (nrows == 0) return;

  unsigned blocks = nrows < 2048u ? nrows : 2048u;
  smart_embedding_kernel<<<dim3(blocks), dim3(192), 0, stream>>>(
      price_w, size_w, exchange_w, pair_w, level_w, time_w,
      (v4f*)d_out, nrows);
}